// WeightedRankingLoss_61392262529299
// MI455X (gfx1250) — compile-verified
//
#include <hip/hip_runtime.h>

#define NB   32      // batch rows
#define NN   1024    // row length
#define TPB  256     // threads per block (8 wave32)
#define GY   8       // i-tile groups per row (8 waves * 8 groups = 64 i-tiles)
#define EPS  1e-12f

typedef float v2f __attribute__((ext_vector_type(2)));
typedef float v8f __attribute__((ext_vector_type(8)));

// ---------------------------------------------------------------------------
// Kernel 1: per-row weights (top-5 + quantile tail) and listwise CE loss.
// One block per row. LDS bitonic sort for order statistics.
// ---------------------------------------------------------------------------
__global__ void __launch_bounds__(TPB)
rowstats_kernel(const float* __restrict__ yp, const float* __restrict__ yt,
                float* __restrict__ w_out, float* __restrict__ list_out) {
  __shared__ float s[NN];    // sort buffer, later reused for weights
  __shared__ float tt[NN];   // y_true row
  __shared__ float pp[NN];   // y_pred row
  __shared__ float red[TPB];
  __shared__ float sc[2];    // threshold, kth-largest

  const int b = blockIdx.x;
  const int tid = threadIdx.x;

  for (int i = tid; i < NN; i += TPB) {
    float t = yt[b * NN + i];
    float p = yp[b * NN + i];
    tt[i] = t; pp[i] = p; s[i] = t;
  }
  __syncthreads();

  // Bitonic sort ascending (1024 elements, 256 threads).
  for (int k = 2; k <= NN; k <<= 1) {
    for (int j = k >> 1; j > 0; j >>= 1) {
      for (int i = tid; i < NN; i += TPB) {
        int ixj = i ^ j;
        if (ixj > i) {
          float a = s[i], c = s[ixj];
          bool up = ((i & k) == 0);
          if ((a > c) == up) { s[i] = c; s[ixj] = a; }
        }
      }
      __syncthreads();
    }
  }

  if (tid == 0) {
    // jnp.quantile(0.95, linear): pos = 0.95*(N-1) = 971.85
    float frac = 0.95f * (float)(NN - 1) - 971.0f;
    sc[0] = s[971] + frac * (s[972] - s[971]);  // tail threshold
    sc[1] = s[NN - 5];                          // 5th largest (top-k boundary)
  }
  __syncthreads();
  const float thr = sc[0];
  const float kth = sc[1];
  __syncthreads();

  // Weights: base 1, top-5 -> 2, tail (>= thr) -> x10. Reuse s[] for weights.
  for (int i = tid; i < NN; i += TPB) {
    float t = tt[i];
    float w = (t >= kth) ? 2.0f : 1.0f;
    if (t >= thr) w *= 10.0f;
    s[i] = w;
    w_out[b * NN + i] = w;
  }
  __syncthreads();

  // Row maxima for stable softmax.
  float mp = -3.402823466e38f, mt = -3.402823466e38f;
  for (int i = tid; i < NN; i += TPB) {
    mp = fmaxf(mp, pp[i]); mt = fmaxf(mt, tt[i]);
  }
  red[tid] = mp; __syncthreads();
  for (int st = TPB / 2; st > 0; st >>= 1) {
    if (tid < st) red[tid] = fmaxf(red[tid], red[tid + st]);
    __syncthreads();
  }
  mp = red[0]; __syncthreads();
  red[tid] = mt; __syncthreads();
  for (int st = TPB / 2; st > 0; st >>= 1) {
    if (tid < st) red[tid] = fmaxf(red[tid], red[tid + st]);
    __syncthreads();
  }
  mt = red[0]; __syncthreads();

  // Softmax denominators.
  float sp = 0.0f, st_ = 0.0f;
  for (int i = tid; i < NN; i += TPB) {
    sp  += __expf(pp[i] - mp);
    st_ += __expf(tt[i] - mt);
  }
  red[tid] = sp; __syncthreads();
  for (int st = TPB / 2; st > 0; st >>= 1) {
    if (tid < st) red[tid] += red[tid + st];
    __syncthreads();
  }
  float Zp = red[0]; __syncthreads();
  red[tid] = st_; __syncthreads();
  for (int st = TPB / 2; st > 0; st >>= 1) {
    if (tid < st) red[tid] += red[tid + st];
    __syncthreads();
  }
  float Zt = red[0]; __syncthreads();

  // Weighted cross entropy.
  const float invZp = __builtin_amdgcn_rcpf(Zp);
  const float invZt = __builtin_amdgcn_rcpf(Zt);
  float ce = 0.0f, wsum = 0.0f;
  for (int i = tid; i < NN; i += TPB) {
    float tp = __expf(tt[i] - mt) * invZt;
    float pq = __expf(pp[i] - mp) * invZp;
    float w  = s[i];
    ce   -= tp * __logf(pq + EPS) * w;
    wsum += w;
  }
  red[tid] = ce; __syncthreads();
  for (int st = TPB / 2; st > 0; st >>= 1) {
    if (tid < st) red[tid] += red[tid + st];
    __syncthreads();
  }
  ce = red[0]; __syncthreads();
  red[tid] = wsum; __syncthreads();
  for (int st = TPB / 2; st > 0; st >>= 1) {
    if (tid < st) red[tid] += red[tid + st];
    __syncthreads();
  }
  wsum = red[0];
  if (tid == 0) list_out[b] = ce / (wsum + EPS);
}

// ---------------------------------------------------------------------------
// Kernel 2: pairwise loss partials. Grid (NB, GY), 8 waves/block, one i-tile
// per wave, looping 64 j-tiles. The three rank-2 broadcast matrices
// (pred_diff, true_diff, w_i+w_j) are produced per 16x16 tile by a single
// V_WMMA_F32_16X16X4_F32 each: A rows = [x_i, 1 | 0, 0], B cols = [1, +-x_j |
// 0, 0]. Exact f32; identical C/D layouts so all follow-up math is per-lane.
// ---------------------------------------------------------------------------
__global__ void __launch_bounds__(TPB)
pairwise_kernel(const float* __restrict__ yp, const float* __restrict__ yt,
                const float* __restrict__ w_in, float* __restrict__ part_out) {
  __shared__ float lp[NN], lt[NN], lw[NN];
  __shared__ float redA[TPB], redC[TPB];

  const int b = blockIdx.x;
  const int tid = threadIdx.x;

  for (int i = tid; i < NN; i += TPB) {
    lp[i] = yp[b * NN + i];
    lt[i] = yt[b * NN + i];
    lw[i] = w_in[b * NN + i];
  }
  __syncthreads();

  const int wave  = tid >> 5;       // 0..7
  const int lane  = tid & 31;
  const int laneL = lane & 15;
  const bool lo   = (lane < 16);    // lanes 0-15 carry K=0,1; lanes 16-31 -> 0
  const float one = lo ? 1.0f : 0.0f;

  const int I = blockIdx.y * 8 + wave;        // i-tile 0..63
  const int ib = I * 16 + laneL;
  const float pi = lo ? lp[ib] : 0.0f;
  const float ti = lo ? lt[ib] : 0.0f;
  const float wi = lo ? lw[ib] : 0.0f;
  const v2f aP = {pi, one};
  const v2f aT = {ti, one};
  const v2f aW = {wi, one};
  const v8f cz = {0.f, 0.f, 0.f, 0.f, 0.f, 0.f, 0.f, 0.f};

  float acc = 0.0f, cnt = 0.0f;
  for (int J = 0; J < NN / 16; ++J) {
    const int jb = J * 16 + laneL;
    const float pj = lo ? lp[jb] : 0.0f;
    const float tj = lo ? lt[jb] : 0.0f;
    const float wj = lo ? lw[jb] : 0.0f;
    const v2f bP = {one, -pj};
    const v2f bT = {one, -tj};
    const v2f bW = {one,  wj};

    // (neg_a, A, neg_b, B, c_mod, C, reuse_a, reuse_b)
    v8f PD = __builtin_amdgcn_wmma_f32_16x16x4_f32(false, aP, false, bP,
                                                   (short)0, cz, false, false);
    v8f TD = __builtin_amdgcn_wmma_f32_16x16x4_f32(false, aT, false, bT,
                                                   (short)0, cz, false, false);
    v8f WS = __builtin_amdgcn_wmma_f32_16x16x4_f32(false, aW, false, bW,
                                                   (short)0, cz, false, false);
#pragma unroll
    for (int v = 0; v < 8; ++v) {
      float td = TD[v];
      float m  = (td != 0.0f) ? 1.0f : 0.0f;          // mask (kills diagonal)
      float x  = (td > 0.0f) ? PD[v] : -PD[v];        // pred_diff * sign(true_diff)
      float pw = __builtin_amdgcn_rcpf(1.0f + __expf(x));  // sigmoid(-x)
      acc += pw * WS[v] * m;
      cnt += m;
    }
  }

  redA[tid] = acc; redC[tid] = cnt; __syncthreads();
  for (int st = TPB / 2; st > 0; st >>= 1) {
    if (tid < st) { redA[tid] += redA[tid + st]; redC[tid] += redC[tid + st]; }
    __syncthreads();
  }
  if (tid == 0) {
    const int slot = (b * GY + blockIdx.y) * 2;
    part_out[slot + 0] = redA[0];
    part_out[slot + 1] = redC[0];
  }
}

// ---------------------------------------------------------------------------
// Kernel 3: combine -> scalar. mean_b(listwise_b) + mean_b(pairwise_b).
// ---------------------------------------------------------------------------
__global__ void final_kernel(const float* __restrict__ list_in,
                             const float* __restrict__ part_in,
                             float* __restrict__ out) {
  __shared__ float red[NB];
  const int tid = threadIdx.x;   // 32 threads
  float v = 0.0f;
  if (tid < NB) {
    float acc = 0.0f, cnt = 0.0f;
    for (int g = 0; g < GY; ++g) {
      acc += part_in[(tid * GY + g) * 2 + 0];
      cnt += part_in[(tid * GY + g) * 2 + 1];
    }
    float pr = acc / fmaxf(cnt, 1.0f);
    v = list_in[tid] + pr;
  }
  red[tid] = v; __syncthreads();
  for (int s = NB / 2; s > 0; s >>= 1) {
    if (tid < s) red[tid] += red[tid + s];
    __syncthreads();
  }
  if (tid == 0) out[0] = red[0] * (1.0f / (float)NB);
}

// ---------------------------------------------------------------------------
extern "C" void kernel_launch(void* const* d_in, const int* in_sizes, int n_in,
                              void* d_out, int out_size, void* d_ws, size_t ws_size,
                              hipStream_t stream) {
  const float* yp = (const float*)d_in[0];
  const float* yt = (const float*)d_in[1];
  float* out = (float*)d_out;

  float* ws       = (float*)d_ws;
  float* w_buf    = ws;                // NB*NN weights
  float* list_buf = w_buf + NB * NN;   // NB listwise per-row
  float* part_buf = list_buf + NB;     // NB*GY*2 pairwise partials

  rowstats_kernel<<<NB, TPB, 0, stream>>>(yp, yt, w_buf, list_buf);
  pairwise_kernel<<<dim3(NB, GY), TPB, 0, stream>>>(yp, yt, w_buf, part_buf);
  final_kernel<<<1, 32, 0, stream>>>(list_buf, part_buf, out);
}